// PyramidROIAlign_36687610643036
// MI455X (gfx1250) — compile-verified
//
#include <hip/hip_runtime.h>

// ---------------------------------------------------------------------------
// PyramidROIAlign for MI455X (gfx1250), CDNA5 paths:
//   * global_load_async_to_lds_b128 + s_wait_asynccnt : gather staging DMA
//   * v_wmma_f32_16x16x4_f32                          : x-direction lerp as
//       out[px,c] = sum_j Wx[px,j] * G[j,c]  (Wx 14x28, 2 nonzeros/row)
// Op is HBM-bound (~1 GB traffic / 23.3 TB/s ~= 45 us floor, ~0.4 GFLOP),
// so the matrix pipe is free real estate; loads stay 128-bit coalesced, stores are 32-bit in 64 B runs.
// ---------------------------------------------------------------------------

#define NUM_ROI   1024     // B*N = 2*512
#define POOL      14
#define CH        256
#define CH4       64       // CH/4 float4 chunks
#define NSLOT     56       // 28 x-columns * 2 y-rows staged per (roi, py)

typedef float v2f __attribute__((ext_vector_type(2)));
typedef float v8f __attribute__((ext_vector_type(8)));

// CDNA5 async copy: 16 bytes global -> LDS per lane, tracked by ASYNCcnt.
__device__ __forceinline__ void async_ld_b128(unsigned lds_byte_off, const float* g) {
    asm volatile("global_load_async_to_lds_b128 %0, %1, off"
                 :: "v"(lds_byte_off),
                    "v"((unsigned long long)(size_t)g)
                 : "memory");
}

__device__ __forceinline__ void wait_async0() {
    asm volatile("s_wait_asynccnt 0x0" ::: "memory");
}

__device__ __forceinline__ int clampi(int v, int lo, int hi) {
    return v < lo ? lo : (v > hi ? hi : v);
}

__global__ __launch_bounds__(256)
void roi_align_gfx1250(const float* __restrict__ boxes,
                       const float* __restrict__ p2,
                       const float* __restrict__ p3,
                       const float* __restrict__ p4,
                       const float* __restrict__ p5,
                       float* __restrict__ out)
{
    // 56 column-vectors of 256 ch staged by the async DMA path: 56 KB.
    __shared__ float smem[NSLOT * CH];
    __shared__ int   s_col[2 * POOL];   // per-px source columns (x0, x1)
    __shared__ float s_lx[POOL];        // per-px x lerp weight

    const int blk = blockIdx.x;          // 0 .. NUM_ROI*POOL-1
    const int roi = blk / POOL;
    const int py  = blk - roi * POOL;
    const int tid = threadIdx.x;
    const int b   = roi >> 9;            // batch index (N = 512)

    // Box (normalized y1,x1,y2,x2) — uniform across block, scalar-promoted.
    const float by1 = boxes[roi * 4 + 0];
    const float bx1 = boxes[roi * 4 + 1];
    const float by2 = boxes[roi * 4 + 2];
    const float bx2 = boxes[roi * 4 + 3];
    const float bh  = by2 - by1;
    const float bw  = bx2 - bx1;

    // roi_level = clip(4 + round(log2(sqrt(h*w) / (224/1024))), 2, 5)
    float rl = 4.0f + rintf(log2f(sqrtf(bh * bw) / 0.21875f));
    rl = fminf(fmaxf(rl, 2.0f), 5.0f);   // handles -inf (zero-area) -> level 2
    const int lvl = (int)rl;

    const float* fm; int H;
    if      (lvl == 2) { fm = p2; H = 256; }
    else if (lvl == 3) { fm = p3; H = 128; }
    else if (lvl == 4) { fm = p4; H = 64;  }
    else               { fm = p5; H = 32;  }
    const int W = H;

    // y sampling for this pooled row (uniform across block).
    const float ty  = (float)py / 13.0f;
    const float ys  = (by1 + ty * bh) * (float)(H - 1);
    const float y0f = floorf(ys);
    const float ly  = ys - y0f;
    const int   iy0 = clampi((int)y0f, 0, H - 1);
    const int   iy1 = (iy0 + 1 > H - 1) ? (H - 1) : (iy0 + 1);
    const float wy0 = 1.0f - ly;
    const float wy1 = ly;

    // Per-px x coordinates computed once (single FP sequence everywhere).
    if (tid < POOL) {
        float xs  = (bx1 + ((float)tid / 13.0f) * bw) * (float)(W - 1);
        float x0f = floorf(xs);
        int   ix0 = clampi((int)x0f, 0, W - 1);
        s_col[2 * tid + 0] = ix0;
        s_col[2 * tid + 1] = (ix0 + 1 > W - 1) ? (W - 1) : (ix0 + 1);
        s_lx[tid] = xs - x0f;
    }
    __syncthreads();

    // ---- Stage 56 KB (28 x-cols x 2 y-rows x 256 ch) via async global->LDS.
    // Slot s: xcol = s>>1 (0..27), yrow = s&1. Each lane moves 16 B;
    // consecutive lanes carry consecutive channels -> coalesced 512 B/wave.
    const unsigned sbase = (unsigned)(size_t)(void*)smem;   // LDS byte offset
    #pragma unroll
    for (int it = 0; it < 14; ++it) {
        int idx  = it * 256 + tid;        // 0 .. 3583
        int s    = idx >> 6;              // slot 0..55
        int c4   = idx & 63;
        int xcol = s >> 1;
        int r    = s & 1;
        int col  = s_col[xcol];
        int row  = r ? iy1 : iy0;
        const float* g = fm + (((size_t)(b * H + row)) * W + col) * CH + c4 * 4;
        async_ld_b128(sbase + (unsigned)(s * CH + c4 * 4) * 4u, g);
    }
    wait_async0();       // this wave's DMAs done (ASYNCcnt == 0)
    __syncthreads();     // all waves' staged data visible

    // ---- Phase A: y-lerp in place. G[xcol][c] = row0*wy0 + row1*wy1,
    // written over the even slot. Each (xcol,c) element is read and written
    // by exactly one thread -> no cross-thread hazard.
    {
        const float4* sm4r = (const float4*)smem;
        float4*       sm4w = (float4*)smem;
        #pragma unroll
        for (int it = 0; it < 7; ++it) {          // 28*64 / 256
            int i    = it * 256 + tid;
            int xcol = i >> 6;
            int c4   = i & 63;
            float4 v0 = sm4r[(2 * xcol + 0) * CH4 + c4];
            float4 v1 = sm4r[(2 * xcol + 1) * CH4 + c4];
            float4 gv;
            gv.x = v0.x * wy0 + v1.x * wy1;
            gv.y = v0.y * wy0 + v1.y * wy1;
            gv.z = v0.z * wy0 + v1.z * wy1;
            gv.w = v0.w * wy0 + v1.w * wy1;
            sm4w[(2 * xcol) * CH4 + c4] = gv;
        }
    }
    __syncthreads();

    // ---- Phase B: x-lerp on the matrix pipe.
    // D(16px x 16ch) += A(16x4) x B(4x16) over 7 K-chunks covering j=0..27.
    // Layouts per CDNA5 ISA 7.12.2 (f32):
    //   A: lane%16 = M(px); lanes<16 hold K={4g,4g+1} in VGPR{0,1},
    //      lanes>=16 hold K={4g+2,4g+3}.
    //   B: lane%16 = N(ch tile col); same lane-half/VGPR -> K mapping.
    //   D: lane%16 = N; VGPR v -> M = v + (lane<16 ? 0 : 8).
    {
        const int wave = tid >> 5;          // wave32
        const int lane = tid & 31;
        const int n    = lane & 15;         // A: px row / B: channel col
        const int kb   = (lane < 16) ? 0 : 2;

        const float lxm    = s_lx[n < 14 ? n : 13];
        const float w_even = (n < 14) ? (1.0f - lxm) : 0.0f;  // Wx[n][2n]
        const float w_odd  = (n < 14) ? lxm          : 0.0f;  // Wx[n][2n+1]
        const int   j_even = 2 * n;
        const int   j_odd  = 2 * n + 1;

        const size_t obase = ((size_t)roi * (POOL * POOL) + (size_t)py * POOL) * CH;
        const int    px0   = (lane < 16) ? 0 : 8;

        #pragma unroll
        for (int t = 0; t < 2; ++t) {       // 8 waves x 2 tiles = 16 ch tiles
            const int n0 = (wave * 2 + t) * 16;
            v8f acc = {};
            #pragma unroll
            for (int g = 0; g < 7; ++g) {
                const int j0 = 4 * g + kb;
                const int j1 = j0 + 1;
                v2f a;
                a.x = (j0 == j_even) ? w_even : ((j0 == j_odd) ? w_odd : 0.0f);
                a.y = (j1 == j_even) ? w_even : ((j1 == j_odd) ? w_odd : 0.0f);
                v2f bv;
                bv.x = smem[(2 * j0) * CH + n0 + n];   // G[j0][n0+n]
                bv.y = smem[(2 * j1) * CH + n0 + n];   // G[j1][n0+n]
                acc = __builtin_amdgcn_wmma_f32_16x16x4_f32(
                        false, a, false, bv, (short)0, acc, false, false);
            }
            #pragma unroll
            for (int v = 0; v < 8; ++v) {
                const int px = px0 + v;
                if (px < POOL)
                    out[obase + (size_t)px * CH + n0 + n] = acc[v];
            }
        }
    }
}

extern "C" void kernel_launch(void* const* d_in, const int* in_sizes, int n_in,
                              void* d_out, int out_size, void* d_ws, size_t ws_size,
                              hipStream_t stream) {
    (void)in_sizes; (void)n_in; (void)out_size; (void)d_ws; (void)ws_size;
    const float* boxes = (const float*)d_in[0];
    const float* p2    = (const float*)d_in[1];
    const float* p3    = (const float*)d_in[2];
    const float* p4    = (const float*)d_in[3];
    const float* p5    = (const float*)d_in[4];
    float* out = (float*)d_out;

    dim3 grid(NUM_ROI * POOL);   // one block per (roi, pooled row)
    dim3 block(256);             // 8 wave32s
    roi_align_gfx1250<<<grid, block, 0, stream>>>(boxes, p2, p3, p4, p5, out);
}